// twoTrackNetwork_13657996001326
// MI455X (gfx1250) — compile-verified
//
#include <hip/hip_runtime.h>

#define NN 50000
#define NE 800000
#define NETOT (NE + NN)
#define NG 512

typedef __attribute__((ext_vector_type(2))) float v2f;
typedef __attribute__((ext_vector_type(8))) float v8f;

#define FLAG_RELU_IN  1
#define FLAG_RELU_OUT 2

// ---------------------------------------------------------------------------
// FP32 WMMA GEMM: D[N x M] = op(A[N x K]) @ W[K x M] (+ bias).
// One wave computes a 16 x (16*CT) output strip. Per k-step: one A-fragment
// load + CT B-fragment loads issued as a clause, then CT back-to-back
// V_WMMA_F32_16X16X4_F32 ops. Pointer-increment addressing keeps 64-bit
// multiplies out of the hot loop. N%16==0, M%(16*CT)==0, K%4==0.
// ---------------------------------------------------------------------------
template<int CT>
__global__ void gemm_wmma_f32_t(const float* __restrict__ A, const float* __restrict__ W,
                                const float* __restrict__ bias, float* __restrict__ D,
                                int N, int K, int M, int ldD, int flags)
{
  const int groupsM = (M >> 4) / CT;
  const int nTiles  = (N >> 4) * groupsM;
  int wave = blockIdx.x * blockDim.y + threadIdx.y;
  if (wave >= nTiles) return;                      // wave-uniform exit (EXEC stays full)
  int rowTile  = wave / groupsM;
  int colGroup = wave - rowTile * groupsM;
  int r0 = rowTile << 4;
  int c0 = (colGroup * CT) << 4;
  int lane = threadIdx.x;
  int half = lane >> 4;                            // 0: lanes 0-15, 1: lanes 16-31
  int l15  = lane & 15;

  v8f acc[CT];
  #pragma unroll
  for (int t = 0; t < CT; ++t) acc[t] = (v8f){0.f,0.f,0.f,0.f,0.f,0.f,0.f,0.f};

  // A: lane holds one M row; K pair {0,1} (lanes 0-15) or {2,3} (lanes 16-31)
  const float* ap = A + (size_t)(r0 + l15) * K + (half << 1);
  // B: row striped across lanes; rows ka, ka+1
  const float* wp = W + (size_t)(half << 1) * M + c0 + l15;
  const size_t wstep = (size_t)4 * M;
  const bool relu_in = (flags & FLAG_RELU_IN) != 0;

  for (int k0 = 0; k0 < K; k0 += 4) {
    v2f a;
    a.x = ap[0]; a.y = ap[1];
    if (relu_in) { a.x = fmaxf(a.x, 0.f); a.y = fmaxf(a.y, 0.f); }
    v2f b[CT];
    #pragma unroll
    for (int t = 0; t < CT; ++t) {                 // issue all B loads as one clause
      b[t].x = wp[t * 16];
      b[t].y = wp[t * 16 + M];
    }
    #pragma unroll
    for (int t = 0; t < CT; ++t) {                 // then CT uninterrupted WMMAs
      acc[t] = __builtin_amdgcn_wmma_f32_16x16x4_f32(false, a, false, b[t], (short)0,
                                                     acc[t], false, false);
    }
    ap += 4;
    wp += wstep;
  }

  #pragma unroll
  for (int t = 0; t < CT; ++t) {
    int col = c0 + t * 16 + l15;
    float bv = bias ? bias[col] : 0.f;
    float* drow = D + (size_t)(r0 + (half << 3)) * ldD + col;  // C/D: VGPR j -> row j(+8)
    #pragma unroll
    for (int j = 0; j < 8; ++j) {
      float v = acc[t][j] + bv;
      if (flags & FLAG_RELU_OUT) v = fmaxf(v, 0.f);
      drow[(size_t)j * ldD] = v;
    }
  }
}

// ---------------------------------------------------------------------------
// Elementwise / utility kernels
// ---------------------------------------------------------------------------
__global__ void k_fill(float* __restrict__ p, float v, int n) {
  int i = blockIdx.x * blockDim.x + threadIdx.x;
  if (i < n) p[i] = v;
}

__global__ void k_init_bias(float* __restrict__ out, const float* __restrict__ bias,
                            int C, int n) {
  int i = blockIdx.x * blockDim.x + threadIdx.x;
  if (i < n) out[i] = bias[i % C];
}

__global__ void k_add_inplace(float* __restrict__ a, const float* __restrict__ b, int n) {
  int i = blockIdx.x * blockDim.x + threadIdx.x;
  if (i < n) a[i] += b[i];
}

__device__ __forceinline__ void atomicMaxFloat(float* addr, float val) {
  if (val >= 0.0f) atomicMax((int*)addr, __float_as_int(val));
  else             atomicMin((unsigned int*)addr, __float_as_uint(val));
}

__device__ __forceinline__ void edge_sd(int e, const int* __restrict__ src,
                                        const int* __restrict__ dst, int& s_, int& d_) {
  if (e < NE) { s_ = src[e]; d_ = dst[e]; }
  else        { s_ = d_ = e - NE; }                // implicit self-loops
}

// ---------------------------------------------------------------------------
// GAT conv pieces
// ---------------------------------------------------------------------------
// Wave-per-node attention projection: lanes sweep channels (coalesced row
// reads), then a 5-step shuffle tree reduces across the wave32.
__global__ void k_att_proj(const float* __restrict__ z, const float* __restrict__ asw,
                           const float* __restrict__ adw, float* __restrict__ a_s,
                           float* __restrict__ a_d, int C) {
  int wavesPerBlock = blockDim.x >> 5;
  int n = blockIdx.x * wavesPerBlock + (threadIdx.x >> 5);
  if (n >= NN) return;
  int lane = threadIdx.x & 31;
  const float* zr = z + (size_t)n * C;
  float s = 0.f, d = 0.f;
  for (int c = lane; c < C; c += 32) {
    float v = zr[c]; s += v * asw[c]; d += v * adw[c];
  }
  #pragma unroll
  for (int o = 16; o >= 1; o >>= 1) {
    s += __shfl_xor(s, o, 32);
    d += __shfl_xor(d, o, 32);
  }
  if (lane == 0) { a_s[n] = s; a_d[n] = d; }
}

__global__ void k_edge_max(const float* __restrict__ a_s, const float* __restrict__ a_d,
                           const int* __restrict__ src, const int* __restrict__ dst,
                           float* __restrict__ m) {
  int e = blockIdx.x * blockDim.x + threadIdx.x;
  if (e >= NETOT) return;
  int s_, d_; edge_sd(e, src, dst, s_, d_);
  float v = a_s[s_] + a_d[d_];
  v = v > 0.f ? v : 0.2f * v;                      // leaky_relu(0.2)
  atomicMaxFloat(&m[d_], v);
}

__global__ void k_edge_expsum(const float* __restrict__ a_s, const float* __restrict__ a_d,
                              const int* __restrict__ src, const int* __restrict__ dst,
                              const float* __restrict__ m, float* __restrict__ sb,
                              float* __restrict__ pb) {
  int e = blockIdx.x * blockDim.x + threadIdx.x;
  if (e >= NETOT) return;
  int s_, d_; edge_sd(e, src, dst, s_, d_);
  float v = a_s[s_] + a_d[d_];
  v = v > 0.f ? v : 0.2f * v;
  float p = expf(v - m[d_]);
  pb[e] = p;
  atomicAdd(&sb[d_], p);
}

__global__ void k_edge_alpha(float* __restrict__ pb, const float* __restrict__ sb,
                             const int* __restrict__ src, const int* __restrict__ dst) {
  int e = blockIdx.x * blockDim.x + threadIdx.x;
  if (e >= NETOT) return;
  int s_, d_; edge_sd(e, src, dst, s_, d_);
  pb[e] = pb[e] / sb[d_];
}

__global__ void k_edge_scatter(const float* __restrict__ z, const float* __restrict__ pb,
                               const int* __restrict__ src, const int* __restrict__ dst,
                               float* __restrict__ out, int C, unsigned total) {
  unsigned i = blockIdx.x * blockDim.x + threadIdx.x;
  if (i >= total) return;
  unsigned e = i / (unsigned)C;
  int c = (int)(i - e * (unsigned)C);
  int s_, d_; edge_sd((int)e, src, dst, s_, d_);
  atomicAdd(&out[(size_t)d_ * C + c], pb[e] * z[(size_t)s_ * C + c]);
}

// ---------------------------------------------------------------------------
// BatchNorm (HID=16)
// ---------------------------------------------------------------------------
__global__ void k_bn_stats(const float* __restrict__ h, float* __restrict__ sum,
                           float* __restrict__ sumsq) {
  int t = blockIdx.x * blockDim.x + threadIdx.x;
  int c = t & 15;
  int strideN = (gridDim.x * blockDim.x) >> 4;
  float a = 0.f, a2 = 0.f;
  for (int n = t >> 4; n < NN; n += strideN) {
    float v = h[(size_t)n * 16 + c]; a += v; a2 += v * v;
  }
  atomicAdd(&sum[c], a); atomicAdd(&sumsq[c], a2);
}

__global__ void k_bn_finalize(const float* __restrict__ sum, const float* __restrict__ sumsq,
                              const float* __restrict__ gamma, const float* __restrict__ beta,
                              float* __restrict__ scale, float* __restrict__ shift) {
  int c = threadIdx.x;
  if (c >= 16) return;
  float mu  = sum[c] / (float)NN;
  float var = sumsq[c] / (float)NN - mu * mu;
  float sc  = gamma[c] * rsqrtf(var + 1e-5f);
  scale[c] = sc; shift[c] = beta[c] - mu * sc;
}

__global__ void k_bn_apply_relu(const float* __restrict__ h, const float* __restrict__ scale,
                                const float* __restrict__ shift, float* __restrict__ out) {
  int i = blockIdx.x * blockDim.x + threadIdx.x;
  if (i >= NN * 16) return;
  int c = i & 15;
  out[i] = fmaxf(h[i] * scale[c] + shift[c], 0.f);
}

// ---------------------------------------------------------------------------
// MPNN scatter, pooling, final head
// ---------------------------------------------------------------------------
__global__ void k_mpnn_scatter(const float* __restrict__ hx, const int* __restrict__ src,
                               const int* __restrict__ dst, float* __restrict__ out,
                               unsigned total) {
  unsigned i = blockIdx.x * blockDim.x + threadIdx.x;
  if (i >= total) return;
  unsigned e = i >> 7;                             // C = 128
  int c = (int)(i & 127u);
  atomicAdd(&out[(size_t)dst[e] * 128 + c], hx[(size_t)src[e] * 128 + c]);
}

__global__ void k_count(const int* __restrict__ batch, float* __restrict__ cnt) {
  int n = blockIdx.x * blockDim.x + threadIdx.x;
  if (n < NN) atomicAdd(&cnt[batch[n]], 1.0f);
}

__global__ void k_pool_sum(const float* __restrict__ h, const int* __restrict__ batch,
                           float* __restrict__ ps) {
  int i = blockIdx.x * blockDim.x + threadIdx.x;
  if (i >= NN * 128) return;
  int n = i >> 7, c = i & 127;
  atomicAdd(&ps[(size_t)batch[n] * 128 + c], h[i]);
}

__global__ void k_pool_write(const float* __restrict__ ps, const float* __restrict__ cnt,
                             float* __restrict__ outcat, int off) {
  int i = blockIdx.x * blockDim.x + threadIdx.x;
  if (i >= NG * 128) return;
  int g = i >> 7, c = i & 127;
  outcat[(size_t)g * 512 + off + c] = ps[i] / fmaxf(cnt[g], 1.0f);
}

__global__ void k_final(const float* __restrict__ a1, const float* __restrict__ w2,
                        const float* __restrict__ b2, float* __restrict__ out) {
  int g = blockIdx.x * blockDim.x + threadIdx.x;
  if (g >= NG) return;
  float acc = b2[0];
  const float* r = a1 + (size_t)g * 128;
  for (int c = 0; c < 128; ++c) acc += r[c] * w2[c];
  out[g] = acc;
}

// ---------------------------------------------------------------------------
// Host-side launchers
// ---------------------------------------------------------------------------
static inline void gemm(const float* A, const float* W, const float* bias, float* D,
                        int N, int K, int M, int ldD, int flags, hipStream_t st) {
  int tilesM = M / 16;
  dim3 blk(32, 4);
  if (tilesM % 4 == 0) {
    int nTiles = (N / 16) * (tilesM / 4);
    gemm_wmma_f32_t<4><<<(nTiles + 3) / 4, blk, 0, st>>>(A, W, bias, D, N, K, M, ldD, flags);
  } else if (tilesM % 2 == 0) {
    int nTiles = (N / 16) * (tilesM / 2);
    gemm_wmma_f32_t<2><<<(nTiles + 3) / 4, blk, 0, st>>>(A, W, bias, D, N, K, M, ldD, flags);
  } else {
    int nTiles = (N / 16) * tilesM;
    gemm_wmma_f32_t<1><<<(nTiles + 3) / 4, blk, 0, st>>>(A, W, bias, D, N, K, M, ldD, flags);
  }
}

static void gat_conv(const float* z, const float* asw, const float* adw, const float* bias,
                     int C, const int* src0, const int* dst0, float* out,
                     float* a_s, float* a_d, float* mb, float* sb, float* pb,
                     hipStream_t st) {
  const int B = 256;
  k_att_proj<<<(NN + 7) / 8, B, 0, st>>>(z, asw, adw, a_s, a_d, C);  // 8 waves/block
  k_fill<<<(NN + B - 1) / B, B, 0, st>>>(mb, -__builtin_inff(), NN);
  k_fill<<<(NN + B - 1) / B, B, 0, st>>>(sb, 0.f, NN);
  k_init_bias<<<(NN * C + B - 1) / B, B, 0, st>>>(out, bias, C, NN * C);
  k_edge_max<<<(NETOT + B - 1) / B, B, 0, st>>>(a_s, a_d, src0, dst0, mb);
  k_edge_expsum<<<(NETOT + B - 1) / B, B, 0, st>>>(a_s, a_d, src0, dst0, mb, sb, pb);
  k_edge_alpha<<<(NETOT + B - 1) / B, B, 0, st>>>(pb, sb, src0, dst0);
  unsigned total = (unsigned)NETOT * (unsigned)C;
  k_edge_scatter<<<(total + B - 1) / B, B, 0, st>>>(z, pb, src0, dst0, out, C, total);
}

extern "C" void kernel_launch(void* const* d_in, const int* in_sizes, int n_in,
                              void* d_out, int out_size, void* d_ws, size_t ws_size,
                              hipStream_t stream) {
  (void)in_sizes; (void)n_in; (void)out_size; (void)ws_size;
  const int B = 256;

  const float* x     = (const float*)d_in[0];
  const int*   eidx  = (const int*)d_in[1];
  const int*   src0  = eidx;
  const int*   dst0  = eidx + NE;
  const int*   batch = (const int*)d_in[2];
  const float* clsE  = (const float*)d_in[3];
  const float* meanE = (const float*)d_in[4];
  int i = 5;
  const float *g1W=(const float*)d_in[i++], *g1as=(const float*)d_in[i++],
              *g1ad=(const float*)d_in[i++], *g1b=(const float*)d_in[i++];
  const float *g2W=(const float*)d_in[i++], *g2as=(const float*)d_in[i++],
              *g2ad=(const float*)d_in[i++], *g2b=(const float*)d_in[i++];
  const float *diW=(const float*)d_in[i++], *dias=(const float*)d_in[i++],
              *diad=(const float*)d_in[i++], *dib=(const float*)d_in[i++];
  const float *dmW=(const float*)d_in[i++], *dmas=(const float*)d_in[i++],
              *dmad=(const float*)d_in[i++], *dmb=(const float*)d_in[i++];
  const float *bng=(const float*)d_in[i++], *bnb=(const float*)d_in[i++];
  const float *doW=(const float*)d_in[i++], *doas=(const float*)d_in[i++],
              *doad=(const float*)d_in[i++], *dob=(const float*)d_in[i++];
  const float *mpW=(const float*)d_in[i++],  *mpB=(const float*)d_in[i++];
  const float *mp2W=(const float*)d_in[i++], *mp2B=(const float*)d_in[i++];
  const float *c1W=(const float*)d_in[i++],  *c1B=(const float*)d_in[i++];
  const float *c2W=(const float*)d_in[i++],  *c2B=(const float*)d_in[i++];
  const float *m1W=(const float*)d_in[i++],  *m1B=(const float*)d_in[i++];
  const float *m2W=(const float*)d_in[i++],  *m2B=(const float*)d_in[i++];
  const float *f1W=(const float*)d_in[i++],  *f1B=(const float*)d_in[i++];
  const float *f2W=(const float*)d_in[i++],  *f2B=(const float*)d_in[i++];
  const float *a1W=(const float*)d_in[i++],  *a1B=(const float*)d_in[i++];
  const float *a2W=(const float*)d_in[i++],  *a2B=(const float*)d_in[i++];

  // ---- workspace carve (floats) ----
  float* ws = (float*)d_ws;
  size_t off = 0;
  auto alloc = [&](size_t n) { float* p = ws + off; off += n; return p; };
  float* z16a  = alloc((size_t)NN * 16);
  float* z16b  = alloc((size_t)NN * 16);
  float* h0b   = alloc((size_t)NN * 16);
  float* C1    = alloc((size_t)NN * 128);
  float* C2    = alloc((size_t)NN * 128);
  float* a_s   = alloc(NN);
  float* a_d   = alloc(NN);
  float* mb    = alloc(NN);
  float* sb    = alloc(NN);
  float* pb    = alloc(NETOT);
  float* bnsum = alloc(16);
  float* bnsq  = alloc(16);
  float* bnsc  = alloc(16);
  float* bnsh  = alloc(16);
  float* cnt   = alloc(NG);
  float* psums = alloc((size_t)NG * 128);
  float* ocat  = alloc((size_t)NG * 512);
  float* cm    = alloc((size_t)NG * 128);
  float* t128  = alloc((size_t)NG * 128);
  float* t32   = alloc((size_t)NG * 32);

  // graph node counts (shared by all pools)
  k_fill<<<(NG + B - 1) / B, B, 0, stream>>>(cnt, 0.f, NG);
  k_count<<<(NN + B - 1) / B, B, 0, stream>>>(batch, cnt);

  // ---- GAT branch: conv1 -> relu -> conv2 ----
  gemm(x, g1W, nullptr, z16a, NN, 256, 16, 16, 0, stream);
  gat_conv(z16a, g1as, g1ad, g1b, 16, src0, dst0, z16b, a_s, a_d, mb, sb, pb, stream);
  gemm(z16b, g2W, nullptr, C1, NN, 16, 128, 128, FLAG_RELU_IN, stream);
  gat_conv(C1, g2as, g2ad, g2b, 128, src0, dst0, C2, a_s, a_d, mb, sb, pb, stream);
  k_fill<<<(NG * 128 + B - 1) / B, B, 0, stream>>>(psums, 0.f, NG * 128);
  k_pool_sum<<<(NN * 128 + B - 1) / B, B, 0, stream>>>(C2, batch, psums);
  k_pool_write<<<(NG * 128 + B - 1) / B, B, 0, stream>>>(psums, cnt, ocat, 0);

  // ---- DeepGAT branch ----
  gemm(x, diW, nullptr, z16a, NN, 256, 16, 16, 0, stream);
  gat_conv(z16a, dias, diad, dib, 16, src0, dst0, h0b, a_s, a_d, mb, sb, pb, stream);
  k_fill<<<1, 64, 0, stream>>>(bnsum, 0.f, 32);              // bnsum+bnsq contiguous
  k_bn_stats<<<64, 256, 0, stream>>>(h0b, bnsum, bnsq);
  k_bn_finalize<<<1, 16, 0, stream>>>(bnsum, bnsq, bng, bnb, bnsc, bnsh);
  k_bn_apply_relu<<<(NN * 16 + B - 1) / B, B, 0, stream>>>(h0b, bnsc, bnsh, z16a);
  gemm(z16a, dmW, nullptr, z16b, NN, 16, 16, 16, 0, stream);
  gat_conv(z16b, dmas, dmad, dmb, 16, src0, dst0, z16a, a_s, a_d, mb, sb, pb, stream);
  k_add_inplace<<<(NN * 16 + B - 1) / B, B, 0, stream>>>(h0b, z16a, NN * 16);  // h = h0 + conv
  gemm(h0b, doW, nullptr, C1, NN, 16, 128, 128, 0, stream);
  gat_conv(C1, doas, doad, dob, 128, src0, dst0, C2, a_s, a_d, mb, sb, pb, stream);
  k_fill<<<(NG * 128 + B - 1) / B, B, 0, stream>>>(psums, 0.f, NG * 128);
  k_pool_sum<<<(NN * 128 + B - 1) / B, B, 0, stream>>>(C2, batch, psums);
  k_pool_write<<<(NG * 128 + B - 1) / B, B, 0, stream>>>(psums, cnt, ocat, 128);

  // ---- MPNN branch ----
  gemm(x, mpW, mpB, C1, NN, 256, 128, 128, 0, stream);
  k_fill<<<(NN * 128 + B - 1) / B, B, 0, stream>>>(C2, 0.f, NN * 128);
  {
    unsigned total = (unsigned)NE * 128u;
    k_mpnn_scatter<<<(total + B - 1) / B, B, 0, stream>>>(C1, src0, dst0, C2, total);
  }
  gemm(C2, mp2W, mp2B, C1, NN, 128, 128, 128, 0, stream);
  k_fill<<<(NG * 128 + B - 1) / B, B, 0, stream>>>(psums, 0.f, NG * 128);
  k_pool_sum<<<(NN * 128 + B - 1) / B, B, 0, stream>>>(C1, batch, psums);
  k_pool_write<<<(NG * 128 + B - 1) / B, B, 0, stream>>>(psums, cnt, ocat, 256);

  // ---- MLP branch (512 rows, all WMMA) ----
  gemm(clsE,  c1W, c1B, t128,       NG, 768, 128, 128, FLAG_RELU_OUT, stream);
  gemm(t128,  c2W, c2B, cm,         NG, 128,  64, 128, 0,             stream);  // cm[:, :64]
  gemm(meanE, m1W, m1B, t128,       NG, 768, 128, 128, FLAG_RELU_OUT, stream);
  gemm(t128,  m2W, m2B, cm + 64,    NG, 128,  64, 128, 0,             stream);  // cm[:, 64:]
  gemm(cm,    f1W, f1B, t32,        NG, 128,  32,  32, FLAG_RELU_OUT, stream);
  gemm(t32,   f2W, f2B, ocat + 384, NG, 32,  128, 512, 0,             stream);  // emb cols

  // ---- aggregation head ----
  gemm(ocat, a1W, a1B, t128, NG, 512, 128, 128, FLAG_RELU_OUT, stream);
  k_final<<<(NG + B - 1) / B, B, 0, stream>>>(t128, a2W, a2B, (float*)d_out);
}